// VesselGCN_35966056137295
// MI455X (gfx1250) — compile-verified
//
#include <hip/hip_runtime.h>
#include <hip/hip_bf16.h>

typedef __attribute__((ext_vector_type(2))) float v2f;
typedef __attribute__((ext_vector_type(8))) float v8f;

// Problem constants (B=8, N=256, ND=32, ED=16, H=128, L=3, OUT=1)
#define MROWS 2048   // B*N
#define HDIM  128

// ---------------------------------------------------------------------------
// Masked aggregation of raw edge features over j:
//   EF_agg[row,k] = sum_j adj[row,j] * ef[row,j,k]   (k<16)
//   deg[row]      = sum_j adj[row,j]
// One block (256 threads = 16 groups x 16 features) per (b,i) row.
// This kernel is the bandwidth-dominant pass (~35 MB total).
// ---------------------------------------------------------------------------
__global__ void k_edge_agg(const float* __restrict__ ef,
                           const float* __restrict__ adj,
                           float* __restrict__ ef_agg,
                           float* __restrict__ deg) {
    const int row = blockIdx.x;                 // b*N + i
    const int tid = threadIdx.x;                // 0..255
    const int k = tid & 15;                     // feature index
    const int g = tid >> 4;                     // group 0..15
    const float* __restrict__ efr  = ef  + (size_t)row * 256 * 16;
    const float* __restrict__ adjr = adj + (size_t)row * 256;

    float acc = 0.0f, d = 0.0f;
    for (int j = g; j < 256; j += 16) {
        const float a = adjr[j];
        acc += a * efr[(size_t)j * 16 + k];
        if (k == 0) d += a;
    }

    __shared__ float s[256];
    __shared__ float sd[16];
    s[tid] = acc;
    if (k == 0) sd[g] = d;
    __syncthreads();
    for (int stride = 8; stride > 0; stride >>= 1) {
        if (g < stride) {
            s[g * 16 + k] += s[(g + stride) * 16 + k];
            if (k == 0) sd[g] += sd[g + stride];
        }
        __syncthreads();
    }
    if (g == 0) {
        ef_agg[(size_t)row * 16 + k] = s[k];
        if (k == 0) deg[row] = sd[0];
    }
}

// ---------------------------------------------------------------------------
// Generic fp32 WMMA GEMM: C[M,N] = A[M,K] @ W[K,N] (+bias1[n]) (+rowscale[m]*bias2[n]) (relu)
// One wave computes a 16x64 strip = 4 independent 16x16 accumulators via
// V_WMMA_F32_16X16X4_F32. Per k-step: 1 shared A-fragment load, 4 B-fragment
// loads, 4 independent WMMAs (4 overlapping XDL accumulation chains).
// Requires N % 64 == 0 (true for all call sites: N in {64, 128}).
// A-frag (16x4): lane&15 = M row; lane>=16 holds K+2 pair.   (ISA 7.12.2)
// B-frag (4x16): lane&15 = N col; lane>=16 holds K+2 pair.
// C/D (16x16):   lane&15 = N col; vgpr v = M row (lane>=16 -> M row 8+v).
// ---------------------------------------------------------------------------
__global__ void k_gemm_wmma(const float* __restrict__ A,
                            const float* __restrict__ W,
                            float* __restrict__ C,
                            int M, int N, int K,
                            int lda, int ldw, int ldc,
                            const float* __restrict__ bias1,
                            const float* __restrict__ bias2,
                            const float* __restrict__ rowscale,
                            int relu_flag) {
    const int lane = threadIdx.x & 31;
    const int wib  = threadIdx.x >> 5;
    const int nstrip = N >> 6;                  // strips of 64 columns
    const int tiles = (M >> 4) * nstrip;
    const int tile = blockIdx.x * (blockDim.x >> 5) + wib;
    if (tile >= tiles) return;                  // whole-wave exit; EXEC stays all-1

    const int tm = tile / nstrip;
    const int tn = tile - tm * nstrip;
    const int M0 = tm << 4;
    const int N0 = tn << 6;

    const int mrow = lane & 15;                 // M row (A) / N col (B,C)
    const int koff = (lane >> 4) << 1;          // 0 or 2

    v8f acc0 = {}, acc1 = {}, acc2 = {}, acc3 = {};
    const float* __restrict__ arow = A + (size_t)(M0 + mrow) * lda + koff;
    const float* __restrict__ wcol = W + N0 + mrow;

    for (int k = 0; k < K; k += 4) {
        v2f a;
        a.x = arow[k + 0];
        a.y = arow[k + 1];
        const float* __restrict__ w0 = wcol + (size_t)(k + koff) * ldw;
        const float* __restrict__ w1 = w0 + ldw;
        v2f b0, b1, b2, b3;
        b0.x = w0[0];  b0.y = w1[0];
        b1.x = w0[16]; b1.y = w1[16];
        b2.x = w0[32]; b2.y = w1[32];
        b3.x = w0[48]; b3.y = w1[48];
        acc0 = __builtin_amdgcn_wmma_f32_16x16x4_f32(false, a, false, b0, (short)0, acc0, false, false);
        acc1 = __builtin_amdgcn_wmma_f32_16x16x4_f32(false, a, false, b1, (short)0, acc1, false, false);
        acc2 = __builtin_amdgcn_wmma_f32_16x16x4_f32(false, a, false, b2, (short)0, acc2, false, false);
        acc3 = __builtin_amdgcn_wmma_f32_16x16x4_f32(false, a, false, b3, (short)0, acc3, false, false);
    }

    const int rbase = M0 + ((lane >> 4) << 3);
    const float rsc = (bias2 && rowscale) ? 0.0f : 1.0f;  // placeholder, see below
    (void)rsc;
#pragma unroll
    for (int t = 0; t < 4; ++t) {
        const v8f accT = (t == 0) ? acc0 : (t == 1) ? acc1 : (t == 2) ? acc2 : acc3;
        const int col = N0 + t * 16 + mrow;
        const float b1v = bias1 ? bias1[col] : 0.0f;
        const float b2v = bias2 ? bias2[col] : 0.0f;
#pragma unroll
        for (int v = 0; v < 8; ++v) {
            const int row = rbase + v;
            float val = accT[v] + b1v;
            if (bias2) val += (rowscale ? rowscale[row] : 1.0f) * b2v;
            if (relu_flag) val = fmaxf(val, 0.0f);
            C[(size_t)row * ldc + col] = val;
        }
    }
}

// ---------------------------------------------------------------------------
// Per-layer combined biases:
//   bnm[l,n] = bn[l,:] @ Wm[l,:H,n]
//   bc [l,n] = be[l,:] @ Wm[l,H:,n] + bm[l,n]
// grid = L blocks of H threads.
// ---------------------------------------------------------------------------
__global__ void k_combine_bias(const float* __restrict__ Wm,
                               const float* __restrict__ be,
                               const float* __restrict__ bm,
                               const float* __restrict__ bn,
                               float* __restrict__ bc,
                               float* __restrict__ bnm) {
    const int l = blockIdx.x;
    const int n = threadIdx.x;                  // 0..127
    const float* __restrict__ Wm1 = Wm + (size_t)l * 256 * 128;
    const float* __restrict__ Wm2 = Wm1 + 128 * 128;
    float s1 = 0.0f, s2 = 0.0f;
    for (int k = 0; k < 128; ++k) {
        s1 += bn[l * 128 + k] * Wm1[(size_t)k * 128 + n];
        s2 += be[l * 128 + k] * Wm2[(size_t)k * 128 + n];
    }
    bnm[l * 128 + n] = s1;
    bc[l * 128 + n]  = s2 + bm[l * 128 + n];
}

// ---------------------------------------------------------------------------
// Fused epilogue per layer:
//   v = relu(nt + deg*msg_n + msg_e_agg); if residual v += x; x = LN(v)*lnw + lnb
// One block (128 threads = one H row) per node.
// ---------------------------------------------------------------------------
__global__ void k_fuse_ln(float* __restrict__ X,
                          const float* __restrict__ NT,
                          const float* __restrict__ MN,
                          const float* __restrict__ ME,
                          const float* __restrict__ DEG,
                          const float* __restrict__ lnw,
                          const float* __restrict__ lnb,
                          int residual) {
    const int row = blockIdx.x;
    const int h = threadIdx.x;                  // 0..127
    const size_t idx = (size_t)row * HDIM + h;
    const float d = DEG[row];
    float v = NT[idx] + d * MN[idx] + ME[idx];
    v = fmaxf(v, 0.0f);
    if (residual) v += X[idx];

    __shared__ float ssum[HDIM];
    __shared__ float ssq[HDIM];
    ssum[h] = v;
    ssq[h] = v * v;
    __syncthreads();
    for (int s = HDIM / 2; s > 0; s >>= 1) {
        if (h < s) { ssum[h] += ssum[h + s]; ssq[h] += ssq[h + s]; }
        __syncthreads();
    }
    const float mu  = ssum[0] * (1.0f / HDIM);
    const float var = ssq[0] * (1.0f / HDIM) - mu * mu;
    X[idx] = (v - mu) * rsqrtf(var + 1e-5f) * lnw[h] + lnb[h];
}

// Final head: out[m] = h[m,:] @ Wo2 + bo2
__global__ void k_out_head(const float* __restrict__ Hb,
                           const float* __restrict__ Wo2,
                           const float* __restrict__ bo2,
                           float* __restrict__ out) {
    const int m = blockIdx.x * blockDim.x + threadIdx.x;
    if (m >= MROWS) return;
    float s = bo2[0];
    const float* __restrict__ hr = Hb + (size_t)m * 64;
    for (int k = 0; k < 64; ++k) s += hr[k] * Wo2[k];
    out[m] = s;
}

// ---------------------------------------------------------------------------
extern "C" void kernel_launch(void* const* d_in, const int* in_sizes, int n_in,
                              void* d_out, int out_size, void* d_ws, size_t ws_size,
                              hipStream_t stream) {
    (void)in_sizes; (void)n_in; (void)out_size; (void)ws_size;

    const float* node_feats = (const float*)d_in[0];
    const float* edge_feats = (const float*)d_in[1];
    const float* adjacency  = (const float*)d_in[2];
    const float* Wn0 = (const float*)d_in[3];
    const float* bn0 = (const float*)d_in[4];
    const float* We0 = (const float*)d_in[5];
    const float* be0 = (const float*)d_in[6];
    const float* Wn  = (const float*)d_in[7];
    const float* bn  = (const float*)d_in[8];
    const float* We  = (const float*)d_in[9];
    const float* be  = (const float*)d_in[10];
    const float* Wm  = (const float*)d_in[11];
    const float* bm  = (const float*)d_in[12];
    const float* lnw = (const float*)d_in[13];
    const float* lnb = (const float*)d_in[14];
    const float* Wo1 = (const float*)d_in[15];
    const float* bo1 = (const float*)d_in[16];
    const float* Wo2 = (const float*)d_in[17];
    const float* bo2 = (const float*)d_in[18];

    float* ws = (float*)d_ws;
    float* X    = ws;               // 2048*128
    float* NT   = X    + 262144;    // 2048*128
    float* MN   = NT   + 262144;    // 2048*128
    float* ME   = MN   + 262144;    // 2048*128
    float* EAGG = ME   + 262144;    // 2048*128
    float* EFA  = EAGG + 262144;    // 2048*16
    float* DEG  = EFA  + 32768;     // 2048
    float* HBUF = DEG  + 2048;      // 2048*64
    float* WNM  = HBUF + 131072;    // 3*128*128
    float* WC   = WNM  + 49152;     // 3*128*128
    float* BNM  = WC   + 49152;     // 3*128
    float* BC   = BNM  + 384;       // 3*128

    auto gemm = [&](const float* A, const float* W, float* C, int M, int N, int K,
                    int lda, int ldw, int ldc, const float* b1, const float* b2,
                    const float* rs, int relu) {
        const int tiles = (M / 16) * (N / 64);  // one 16x64 strip per wave
        const int blocks = (tiles + 7) / 8;     // 8 waves (256 thr) per block
        k_gemm_wmma<<<blocks, 256, 0, stream>>>(A, W, C, M, N, K, lda, ldw, ldc,
                                                b1, b2, rs, relu);
    };

    // Stage 0: input transforms + edge aggregation
    gemm(node_feats, Wn0, X, MROWS, 128, 32, 32, 128, 128, bn0, nullptr, nullptr, 0);
    k_edge_agg<<<MROWS, 256, 0, stream>>>(edge_feats, adjacency, EFA, DEG);
    // EAGG = EFA @ We0 + deg * be0
    gemm(EFA, We0, EAGG, MROWS, 128, 16, 16, 128, 128, nullptr, be0, DEG, 0);

    // Combined per-layer weights: WNM[l] = Wn[l]@Wm[l,:H], WC[l] = We[l]@Wm[l,H:]
    for (int l = 0; l < 3; ++l) {
        gemm(Wn + l * 16384, Wm + l * 32768,         WNM + l * 16384,
             128, 128, 128, 128, 128, 128, nullptr, nullptr, nullptr, 0);
        gemm(We + l * 16384, Wm + l * 32768 + 16384, WC  + l * 16384,
             128, 128, 128, 128, 128, 128, nullptr, nullptr, nullptr, 0);
    }
    k_combine_bias<<<3, 128, 0, stream>>>(Wm, be, bm, bn, BC, BNM);

    // GCN layers
    for (int l = 0; l < 3; ++l) {
        gemm(X,    Wn  + l * 16384, NT, MROWS, 128, 128, 128, 128, 128,
             bn  + l * 128, nullptr, nullptr, 0);
        gemm(X,    WNM + l * 16384, MN, MROWS, 128, 128, 128, 128, 128,
             BNM + l * 128, nullptr, nullptr, 0);
        gemm(EAGG, WC  + l * 16384, ME, MROWS, 128, 128, 128, 128, 128,
             nullptr, BC + l * 128, DEG, 0);
        k_fuse_ln<<<MROWS, 128, 0, stream>>>(X, NT, MN, ME, DEG,
                                             lnw + l * 128, lnb + l * 128, l > 0 ? 1 : 0);
    }

    // Output MLP
    gemm(X, Wo1, HBUF, MROWS, 64, 128, 128, 64, 64, bo1, nullptr, nullptr, 1);
    k_out_head<<<MROWS / 256, 256, 0, stream>>>(HBUF, Wo2, bo2, (float*)d_out);
}